// SetOnlyCrossAttention_83313775608043
// MI455X (gfx1250) — compile-verified
//
#include <hip/hip_runtime.h>
#include <hip/hip_bf16.h>
#include <stdint.h>
#include <math.h>

#define BB     4
#define TT     2048
#define DD     512
#define SLEN   4096
#define NSETS  127
#define WINSZ  64
#define STRD   32
#define KH     64
#define VOCABN 32000
#define NPAD   128
#define TOPKN  8

typedef __attribute__((ext_vector_type(2))) float v2f;
typedef __attribute__((ext_vector_type(8))) float v8f;

struct HashConsts { int a[KH]; int b[KH]; };

// ---------------------------------------------------------------------------
// Kernel 1: per (b, set): windowed mean of memory_tokens, minhash signature,
// descriptor MLP (tanh), keys = set_states + desc_router.
// ---------------------------------------------------------------------------
__global__ __launch_bounds__(256) void build_keys_kernel(
    const float* __restrict__ memory_tokens, const int* __restrict__ src_ids,
    const float* __restrict__ W_desc, const float* __restrict__ b_desc,
    float* __restrict__ set_states, float* __restrict__ keys, HashConsts hc)
{
  const int bs   = blockIdx.x;
  const int b    = bs / NSETS;
  const int s    = bs % NSETS;
  const int base = s * STRD;
  const int tid  = threadIdx.x;

  __shared__ int   toks[WINSZ];
  __shared__ float descin[KH + 1];

  if (tid < WINSZ) toks[tid] = src_ids[b * SLEN + base + tid];
  __syncthreads();

  if (tid < KH) {
    const int ha = hc.a[tid], hb = hc.b[tid];
    int mn = VOCABN;
    #pragma unroll 8
    for (int w = 0; w < WINSZ; ++w) {
      int h = (ha * toks[w] + hb) % VOCABN;
      mn = min(mn, h);
    }
    descin[tid] = (float)mn / (float)VOCABN;
  }
  if (tid == KH) descin[KH] = log1pf((float)WINSZ);
  __syncthreads();

  #pragma unroll
  for (int j = 0; j < DD / 256; ++j) {
    const int dd = tid + j * 256;
    float ss = 0.f;
    const float* mp = memory_tokens + (size_t)(b * SLEN + base) * DD + dd;
    for (int w = 0; w < WINSZ; ++w) ss += mp[(size_t)w * DD];
    ss *= (1.0f / (float)WINSZ);
    float acc = b_desc[dd];
    for (int i = 0; i <= KH; ++i) acc += descin[i] * W_desc[i * DD + dd];
    const size_t o = (size_t)(b * NSETS + s) * DD + dd;
    set_states[o] = ss;
    keys[o] = ss + tanhf(acc);
  }
}

// ---------------------------------------------------------------------------
// Kernel 1b: one-time LDS-tiled transpose of W_q (512x512) into Wt[n][k] so
// the qgemm B fragments become contiguous b64 loads.
// ---------------------------------------------------------------------------
__global__ __launch_bounds__(256) void transpose_kernel(
    const float* __restrict__ W, float* __restrict__ Wt)
{
  __shared__ float tile[16][17];
  const int tx = threadIdx.x & 15, ty = threadIdx.x >> 4;
  const int bx = blockIdx.x * 16;  // n block
  const int by = blockIdx.y * 16;  // k block
  tile[ty][tx] = W[(size_t)(by + ty) * DD + bx + tx];
  __syncthreads();
  Wt[(size_t)(bx + ty) * DD + by + tx] = tile[tx][ty];
}

// ---------------------------------------------------------------------------
// Kernel 2: q = token_states @ W_q  (M=8192, N=512, K=512), fp32 WMMA.
// One wave per 16x64 output tile, 4 accumulators, K stepped by 4.
// A frag (16x4 f32): lane<16 row=lane holds K=k0,k0+1; lane>=16 holds K=k0+2,k0+3.
// B frag (4x16 f32): same K split per lane half, N = lane&15; sourced from the
// transposed Wt so each fragment is one contiguous b64 load.
// ---------------------------------------------------------------------------
__global__ __launch_bounds__(32) void qgemm_kernel(
    const float* __restrict__ X, const float* __restrict__ Wt, float* __restrict__ Q)
{
  const int lane = threadIdx.x & 31;
  const int r    = lane & 15;
  const int half = lane >> 4;
  const int m0   = blockIdx.x * 16;
  const int n0   = blockIdx.y * 64;
  v8f acc0 = {}, acc1 = {}, acc2 = {}, acc3 = {};
  const float* arow = X + (size_t)(m0 + r) * DD;
  const float* wr0  = Wt + (size_t)(n0 + r) * DD;
  const float* wr1  = wr0 + 16 * DD;
  const float* wr2  = wr0 + 32 * DD;
  const float* wr3  = wr0 + 48 * DD;
  for (int k0 = 0; k0 < DD; k0 += 4) {
    const int kk = k0 + half * 2;
    v2f a  = *(const v2f*)(arow + kk);
    v2f b0 = *(const v2f*)(wr0 + kk);
    v2f b1 = *(const v2f*)(wr1 + kk);
    v2f b2 = *(const v2f*)(wr2 + kk);
    v2f b3 = *(const v2f*)(wr3 + kk);
    acc0 = __builtin_amdgcn_wmma_f32_16x16x4_f32(false, a, false, b0, (short)0, acc0, false, false);
    acc1 = __builtin_amdgcn_wmma_f32_16x16x4_f32(false, a, false, b1, (short)0, acc1, false, false);
    acc2 = __builtin_amdgcn_wmma_f32_16x16x4_f32(false, a, false, b2, (short)0, acc2, false, false);
    acc3 = __builtin_amdgcn_wmma_f32_16x16x4_f32(false, a, false, b3, (short)0, acc3, false, false);
  }
  #pragma unroll
  for (int i = 0; i < 8; ++i) {
    float* row = Q + (size_t)(m0 + half * 8 + i) * DD + n0 + r;
    row[0]  = acc0[i];
    row[16] = acc1[i];
    row[32] = acc2[i];
    row[48] = acc3[i];
  }
}

// ---------------------------------------------------------------------------
// Kernel 3: logits[b,t,s] = dot(q[b,t,:], keys[b,s,:]) / sqrt(512), staged into
// a [B,T,128] buffer with column 127 padded to -inf. B operand is keys^T, which
// is a contiguous b64 read from the keys row (row-major [s,d]).
// ---------------------------------------------------------------------------
__global__ __launch_bounds__(32) void logits_kernel(
    const float* __restrict__ Q, const float* __restrict__ keys,
    float* __restrict__ logitsPad)
{
  const int lane = threadIdx.x & 31;
  const int r    = lane & 15;
  const int half = lane >> 4;
  const int b    = blockIdx.z;
  const int m0   = blockIdx.x * 16;
  const int n0   = blockIdx.y * 64;
  v8f acc0 = {}, acc1 = {}, acc2 = {}, acc3 = {};
  const float* arow  = Q + ((size_t)b * TT + m0 + r) * DD;
  const float* kbase = keys + (size_t)b * NSETS * DD;
  const int nc0 = n0 + r, nc1 = n0 + 16 + r, nc2 = n0 + 32 + r, nc3 = n0 + 48 + r;
  const float* kr0 = kbase + (size_t)(nc0 < NSETS ? nc0 : 0) * DD;
  const float* kr1 = kbase + (size_t)(nc1 < NSETS ? nc1 : 0) * DD;
  const float* kr2 = kbase + (size_t)(nc2 < NSETS ? nc2 : 0) * DD;
  const float* kr3 = kbase + (size_t)(nc3 < NSETS ? nc3 : 0) * DD;
  for (int k0 = 0; k0 < DD; k0 += 4) {
    const int kk = k0 + half * 2;
    v2f a  = *(const v2f*)(arow + kk);
    v2f b0 = *(const v2f*)(kr0 + kk);
    v2f b1 = *(const v2f*)(kr1 + kk);
    v2f b2 = *(const v2f*)(kr2 + kk);
    v2f b3 = *(const v2f*)(kr3 + kk);
    acc0 = __builtin_amdgcn_wmma_f32_16x16x4_f32(false, a, false, b0, (short)0, acc0, false, false);
    acc1 = __builtin_amdgcn_wmma_f32_16x16x4_f32(false, a, false, b1, (short)0, acc1, false, false);
    acc2 = __builtin_amdgcn_wmma_f32_16x16x4_f32(false, a, false, b2, (short)0, acc2, false, false);
    acc3 = __builtin_amdgcn_wmma_f32_16x16x4_f32(false, a, false, b3, (short)0, acc3, false, false);
  }
  const float scale = 0.044194173824159216f; // 1/sqrt(512)
  const float ninf  = -__builtin_inff();
  #pragma unroll
  for (int i = 0; i < 8; ++i) {
    float* row = logitsPad + ((size_t)b * TT + m0 + half * 8 + i) * NPAD;
    row[nc0] = (nc0 < NSETS) ? acc0[i] * scale : ninf;
    row[nc1] = (nc1 < NSETS) ? acc1[i] * scale : ninf;
    row[nc2] = (nc2 < NSETS) ? acc2[i] * scale : ninf;
    row[nc3] = (nc3 < NSETS) ? acc3[i] * scale : ninf;
  }
}

// ---------------------------------------------------------------------------
// Kernel 4: one wave32 per token: top-8 (wave argmax via shfl_xor, stable
// tie-break on lowest index like lax.top_k), softmax over the 8, scatter the
// probs row, and accumulate token_repr = sum p_i * set_states[s_i].
// ---------------------------------------------------------------------------
__global__ __launch_bounds__(256) void topk_kernel(
    const float* __restrict__ logitsPad, const float* __restrict__ set_states,
    float* __restrict__ token_repr, float* __restrict__ probs)
{
  const int lane  = threadIdx.x & 31;
  const int wave  = threadIdx.x >> 5;
  const int token = blockIdx.x * 8 + wave;
  const int b     = token / TT;

  const float* lp = logitsPad + (size_t)token * NPAD;
  const int sbase = lane * 4;
  float v[4];
  #pragma unroll
  for (int j = 0; j < 4; ++j) v[j] = lp[sbase + j];

  float topv[TOPKN]; int topi[TOPKN];
  for (int it = 0; it < TOPKN; ++it) {
    float lm = v[0]; int li = sbase;
    #pragma unroll
    for (int j = 1; j < 4; ++j) if (v[j] > lm) { lm = v[j]; li = sbase + j; }
    #pragma unroll
    for (int off = 16; off > 0; off >>= 1) {
      float ov = __shfl_xor(lm, off, 32);
      int   oi = __shfl_xor(li, off, 32);
      if (ov > lm || (ov == lm && oi < li)) { lm = ov; li = oi; }
    }
    topv[it] = lm; topi[it] = li;
    if ((li >> 2) == lane) v[li & 3] = -__builtin_inff();
  }

  const float mx = topv[0];
  float p[TOPKN]; float sum = 0.f;
  #pragma unroll
  for (int i = 0; i < TOPKN; ++i) { p[i] = expf(topv[i] - mx); sum += p[i]; }
  const float inv = 1.0f / sum;
  #pragma unroll
  for (int i = 0; i < TOPKN; ++i) p[i] *= inv;

  float* pr = probs + (size_t)token * NSETS;
  for (int s = lane; s < NSETS; s += 32) {
    float pv = 0.f;
    #pragma unroll
    for (int i = 0; i < TOPKN; ++i) if (topi[i] == s) pv = p[i];
    pr[s] = pv;
  }

  float acc[16];
  #pragma unroll
  for (int j = 0; j < 16; ++j) acc[j] = 0.f;
  for (int i = 0; i < TOPKN; ++i) {
    const float* row = set_states + (size_t)(b * NSETS + topi[i]) * DD;
    const float w = p[i];
    #pragma unroll
    for (int j = 0; j < 16; ++j) acc[j] += w * row[lane + j * 32];
  }
  float* rp = token_repr + (size_t)token * DD;
  #pragma unroll
  for (int j = 0; j < 16; ++j) rp[lane + j * 32] = acc[j];
}

// ---------------------------------------------------------------------------
// Host: reproduce numpy default_rng(1234) minhash constants.
// SeedSequence(1234) -> PCG64 (XSL-RR 128/64) -> Generator.integers, which for
// a range <= 2^32 uses 32-bit Lemire bounded sampling over pcg64_next32 (with
// the persistent high-word cache).
// ---------------------------------------------------------------------------
static void make_hash_consts(int* A, int* Bc) {
  typedef unsigned __int128 u128;
  uint32_t pool[4];
  uint32_t hc = 0x43b0d7e5u;
  const uint32_t MULT_A = 0x931e8875u, INIT_B = 0x8b51f9ddu, MULT_B = 0x58f38dedu;
  const uint32_t MIX_L = 0xca01f9ddu, MIX_R = 0x4973f715u;
  const uint32_t entropy0 = 1234u;
  auto hashmix = [&](uint32_t v) -> uint32_t {
    v ^= hc; hc *= MULT_A; v *= hc; v ^= v >> 16; return v;
  };
  auto mixf = [&](uint32_t x, uint32_t y) -> uint32_t {
    uint32_t r = MIX_L * x - MIX_R * y; r ^= r >> 16; return r;
  };
  for (int i = 0; i < 4; ++i) pool[i] = hashmix(i == 0 ? entropy0 : 0u);
  for (int i_src = 0; i_src < 4; ++i_src)
    for (int i_dst = 0; i_dst < 4; ++i_dst)
      if (i_src != i_dst) pool[i_dst] = mixf(pool[i_dst], hashmix(pool[i_src]));

  uint32_t hb = INIT_B;
  uint32_t w[8];
  for (int i = 0; i < 8; ++i) {
    uint32_t dv = pool[i % 4];
    dv ^= hb; hb *= MULT_B; dv *= hb; dv ^= dv >> 16;
    w[i] = dv;
  }
  uint64_t s64[4] = {
    (uint64_t)w[0] | ((uint64_t)w[1] << 32),
    (uint64_t)w[2] | ((uint64_t)w[3] << 32),
    (uint64_t)w[4] | ((uint64_t)w[5] << 32),
    (uint64_t)w[6] | ((uint64_t)w[7] << 32),
  };
  const u128 MUL = (((u128)2549297995355413924ULL) << 64) | 4865540595714422341ULL;
  const u128 initstate = (((u128)s64[0]) << 64) | s64[1];
  const u128 inc = (((((u128)s64[2]) << 64) | s64[3]) << 1) | 1;
  u128 state = 0;
  state = state * MUL + inc;
  state += initstate;
  state = state * MUL + inc;

  int has32 = 0; uint32_t cache32 = 0;
  auto next64 = [&]() -> uint64_t {
    state = state * MUL + inc;
    uint64_t hi = (uint64_t)(state >> 64), lo = (uint64_t)state;
    int rot = (int)(state >> 122) & 63;
    uint64_t x = hi ^ lo;
    return (x >> rot) | (x << ((64 - rot) & 63));
  };
  auto next32 = [&]() -> uint32_t {
    if (has32) { has32 = 0; return cache32; }
    uint64_t v = next64();
    has32 = 1; cache32 = (uint32_t)(v >> 32);
    return (uint32_t)v;
  };
  auto lemire32 = [&](uint32_t rng) -> uint32_t { // rng = inclusive width
    const uint32_t rng_excl = rng + 1u;
    uint64_t m = (uint64_t)next32() * (uint64_t)rng_excl;
    uint32_t leftover = (uint32_t)m;
    if (leftover < rng_excl) {
      const uint32_t threshold = (0xFFFFFFFFu - rng) % rng_excl;
      while (leftover < threshold) {
        m = (uint64_t)next32() * (uint64_t)rng_excl;
        leftover = (uint32_t)m;
      }
    }
    return (uint32_t)(m >> 32);
  };
  for (int i = 0; i < KH; ++i) A[i]  = 1 + (int)lemire32(32766u); // [1, 2^15)
  for (int i = 0; i < KH; ++i) Bc[i] =     (int)lemire32(32767u); // [0, 2^15)
}

extern "C" void kernel_launch(void* const* d_in, const int* in_sizes, int n_in,
                              void* d_out, int out_size, void* d_ws, size_t ws_size,
                              hipStream_t stream) {
  const float* token_states  = (const float*)d_in[0]; // [4,2048,512]
  const float* memory_tokens = (const float*)d_in[1]; // [4,4096,512]
  const int*   src_ids       = (const int*)d_in[2];   // [4,4096]
  const float* W_desc        = (const float*)d_in[3]; // [65,512]
  const float* b_desc        = (const float*)d_in[4]; // [512]
  const float* W_q           = (const float*)d_in[5]; // [512,512]

  float* out        = (float*)d_out;
  float* token_repr = out;                               // B*T*512
  float* probs      = out + (size_t)BB * TT * DD;        // B*T*127

  float* ws         = (float*)d_ws;
  float* set_states = ws;                                // B*S*512
  float* keys       = set_states + (size_t)BB * NSETS * DD;
  float* q          = keys + (size_t)BB * NSETS * DD;    // B*T*512
  float* logitsPad  = q + (size_t)BB * TT * DD;          // B*T*128
  float* wq_t       = logitsPad + (size_t)BB * TT * NPAD; // 512*512

  HashConsts hc;
  make_hash_consts(hc.a, hc.b);

  build_keys_kernel<<<BB * NSETS, 256, 0, stream>>>(
      memory_tokens, src_ids, W_desc, b_desc, set_states, keys, hc);
  transpose_kernel<<<dim3(DD / 16, DD / 16), 256, 0, stream>>>(W_q, wq_t);
  qgemm_kernel<<<dim3((BB * TT) / 16, DD / 64), 32, 0, stream>>>(
      token_states, wq_t, q);
  logits_kernel<<<dim3(TT / 16, NPAD / 64, BB), 32, 0, stream>>>(
      q, keys, logitsPad);
  topk_kernel<<<(BB * TT) / 8, 256, 0, stream>>>(
      logitsPad, set_states, token_repr, probs);
}